// GCNLayer_11622181503329
// MI455X (gfx1250) — compile-verified
//
#include <hip/hip_runtime.h>

// R-GCN layer for MI455X (gfx1250, wave32).
//   out = h + (1/R) * sum_r  scatter_mean( (h @ W[r])[src_r], dst_r )
// Phases (all stream-ordered):
//   1. zero counts, count in-degrees per relation      (int atomics)
//   2. out = h                                         (residual init)
//   3. per relation r: GEMM hW = h @ W[r] via V_WMMA_F32_16X16X4_F32,
//      then scatter hW[src]*scale into out             (f32 atomics)
// Workspace: N*128 floats (hW, reused across relations) + R*N ints (counts)
//            = ~52.4 MB for the reference sizes.

typedef __attribute__((ext_vector_type(2))) float v2f;
typedef __attribute__((ext_vector_type(8))) float v8f;

#define D 128          // feature dim (fixed by the reference)
#define DV4 (D / 4)    // 32 float4 per row

// ---------------------------------------------------------------- utilities
__global__ void zero_int_kernel(int* __restrict__ p, int n) {
    int i = blockIdx.x * blockDim.x + threadIdx.x;
    if (i < n) p[i] = 0;
}

__global__ void copy_f32_kernel(const float* __restrict__ a, float* __restrict__ b, int n) {
    int i = blockIdx.x * blockDim.x + threadIdx.x;
    if (i < n) b[i] = a[i];
}

// in-degree histogram: cnts[r*N + dst[r*E + e]] += 1
__global__ void count_kernel(const int* __restrict__ dst, int* __restrict__ cnts,
                             int total, int N, int E) {
    int i = blockIdx.x * blockDim.x + threadIdx.x;
    if (i >= total) return;
    int r = i / E;
    atomicAdd(&cnts[r * N + dst[i]], 1);
}

// ---------------------------------------------------------------- GEMM (fp32 WMMA)
// hW[rowBase:rowBase+16, :] = h[rowBase:rowBase+16, :] @ W   (W is 128x128, row-major K x N)
// block = 128 threads (4 waves). Wave w computes the 16x16 tile at
// colBase = (blockIdx.y*4 + w)*16. A-tile staged in LDS, shared by all waves.
__global__ void gemm_wmma_f32_kernel(const float* __restrict__ h,
                                     const float* __restrict__ W,
                                     float* __restrict__ hW, int N) {
    __shared__ float As[16 * D];  // 8 KB

    const int rowBase = blockIdx.x * 16;

    // stage 16 x 128 A tile (guarded; N=100000 is a multiple of 16 anyway)
    const float4* __restrict__ h4 = (const float4*)h;
    float4* As4 = (float4*)As;
    for (int i = threadIdx.x; i < 16 * DV4; i += blockDim.x) {
        int rr = i / DV4, cc = i - rr * DV4;
        int grow = rowBase + rr;
        As4[i] = (grow < N) ? h4[(size_t)grow * DV4 + cc] : make_float4(0.f, 0.f, 0.f, 0.f);
    }
    __syncthreads();

    const int wave = threadIdx.x >> 5;
    const int lane = threadIdx.x & 31;
    const int half = lane >> 4;   // 0: lanes 0-15, 1: lanes 16-31
    const int lm   = lane & 15;
    const int colBase = (blockIdx.y * 4 + wave) * 16;

    v8f c = {};
    // K loop: 32 x V_WMMA_F32_16X16X4_F32
    for (int k0 = 0; k0 < D; k0 += 4) {
        const int ka = k0 + 2 * half;  // ISA: lanes 16-31 hold K+2
        v2f a, b;
        // A 16x4: lane lm = row M, VGPR0 = K=ka, VGPR1 = K=ka+1
        a[0] = As[lm * D + ka];
        a[1] = As[lm * D + ka + 1];
        // B 4x16: lane lm = col N, VGPR0 = row K=ka, VGPR1 = row K=ka+1
        b[0] = W[(size_t)ka * D + colBase + lm];
        b[1] = W[(size_t)(ka + 1) * D + colBase + lm];
        c = __builtin_amdgcn_wmma_f32_16x16x4_f32(
                /*neg_a=*/false, a, /*neg_b=*/false, b,
                /*c_mod=*/(short)0, c, /*reuse_a=*/false, /*reuse_b=*/false);
    }

    // C 16x16 f32: VGPR v -> M = v + 8*half, N = lm
    for (int v = 0; v < 8; ++v) {
        int grow = rowBase + v + 8 * half;
        if (grow < N) hW[(size_t)grow * D + colBase + lm] = c[v];
    }
}

// ---------------------------------------------------------------- edge scatter
// One wave per edge: 32 lanes x float4 = full 128-float message row.
// out[dst] += hW[src] * invR / max(cnt[dst], 1)
__global__ void scatter_kernel(const float* __restrict__ hW,
                               const int* __restrict__ src,
                               const int* __restrict__ dst,
                               const int* __restrict__ cnts,
                               float* __restrict__ out,
                               int E, float invR) {
    const int slot = blockIdx.x * (blockDim.x >> 5) + (threadIdx.x >> 5);
    const int lane = threadIdx.x & 31;
    if (slot >= E) return;

    const int s = src[slot];
    const int t = dst[slot];
    const int c = cnts[t];
    const float scale = invR / (float)(c > 0 ? c : 1);

    const float4* __restrict__ v4 = (const float4*)(hW + (size_t)s * D);
    float4 v = v4[lane];

    float* o = out + (size_t)t * D + lane * 4;
    atomicAdd(o + 0, v.x * scale);
    atomicAdd(o + 1, v.y * scale);
    atomicAdd(o + 2, v.z * scale);
    atomicAdd(o + 3, v.w * scale);
}

// ---------------------------------------------------------------- launch
extern "C" void kernel_launch(void* const* d_in, const int* in_sizes, int n_in,
                              void* d_out, int out_size, void* d_ws, size_t ws_size,
                              hipStream_t stream) {
    const float* h   = (const float*)d_in[0];   // [N, 128]
    const float* W   = (const float*)d_in[1];   // [R, 128, 128]
    const int*   src = (const int*)d_in[2];     // [R, E]
    const int*   dst = (const int*)d_in[3];     // [R, E]
    float*       out = (float*)d_out;           // [N, 128]

    const int d = D;
    const int N = in_sizes[0] / d;
    const int R = in_sizes[1] / (d * d);
    const int E = in_sizes[2] / R;

    // workspace layout: hW [N,128] f32, then cnts [R,N] i32
    float* hW   = (float*)d_ws;
    int*   cnts = (int*)((char*)d_ws + (size_t)N * d * sizeof(float));

    // 1) in-degree counts (zero first: ws is poisoned, and must be re-zeroed per call)
    {
        int n = R * N;
        zero_int_kernel<<<(n + 255) / 256, 256, 0, stream>>>(cnts, n);
        int total = R * E;
        count_kernel<<<(total + 255) / 256, 256, 0, stream>>>(dst, cnts, total, N, E);
    }

    // 2) residual init: out = h  (mean_r of the +h residual is h)
    {
        int n = N * d;
        copy_f32_kernel<<<(n + 255) / 256, 256, 0, stream>>>(h, out, n);
    }

    // 3) per relation: GEMM then scatter (stream order serializes reuse of hW)
    const float invR = 1.0f / (float)R;
    for (int r = 0; r < R; ++r) {
        dim3 ggrid((N + 15) / 16, d / (16 * 4));   // 6250 x 2 blocks, 4 waves each
        gemm_wmma_f32_kernel<<<ggrid, 128, 0, stream>>>(h, W + (size_t)r * d * d, hW, N);

        dim3 sgrid((E + 7) / 8);                   // 8 edges (waves) per 256-thread block
        scatter_kernel<<<sgrid, 256, 0, stream>>>(hW, src + (size_t)r * E,
                                                  dst + (size_t)r * E,
                                                  cnts + (size_t)r * N,
                                                  out, E, invR);
    }
}